// ROIAlign_4750233829688
// MI455X (gfx1250) — compile-verified
//
#include <hip/hip_runtime.h>

#define TPB   256
#define OUTS  7
#define NBINS 49

struct SampTab { int lo; int hi; float wl; float wh; };

// ---- CDNA5 async gather: per-lane global -> per-lane LDS, tracked by ASYNCcnt ----
__device__ __forceinline__ void async_gather_b32(unsigned lds_byte_addr, const float* gaddr) {
  asm volatile("global_load_async_to_lds_b32 %0, %1, off"
               :: "v"(lds_byte_addr), "v"(gaddr)
               : "memory");
}
__device__ __forceinline__ void wait_async_le16() { asm volatile("s_wait_asynccnt 16" ::: "memory"); }
__device__ __forceinline__ void wait_async_0()    { asm volatile("s_wait_asynccnt 0"  ::: "memory"); }
__device__ __forceinline__ void wait_ds_0()       { asm volatile("s_wait_dscnt 0"     ::: "memory"); }

struct Stage {
  float wy0, wy1, wy2, wy3;   // row weights (2 subsamples x {lo,hi})
  float wx0, wx1, wx2, wx3;   // col weights
  int   p;                    // linear (c*49 + bin)
};

__device__ __forceinline__ Stage issue_stage(
    int j, int tid, const float* base0, int HW, int W,
    const SampTab* rt, const SampTab* ct, unsigned gbuf_base)
{
  Stage st;
  int p   = tid + j * TPB;
  st.p    = p;
  int c   = p / NBINS;
  int bin = p - c * NBINS;
  int oy  = bin / OUTS;
  int ox  = bin - oy * OUTS;

  SampTab r0 = rt[2 * oy], r1 = rt[2 * oy + 1];
  SampTab c0 = ct[2 * ox], c1 = ct[2 * ox + 1];
  st.wy0 = r0.wl; st.wy1 = r0.wh; st.wy2 = r1.wl; st.wy3 = r1.wh;
  st.wx0 = c0.wl; st.wx1 = c0.wh; st.wx2 = c1.wl; st.wx3 = c1.wh;

  const float* basec = base0 + (size_t)c * (size_t)HW;
  int rofs[4] = { r0.lo * W, r0.hi * W, r1.lo * W, r1.hi * W };
  int cofs[4] = { c0.lo, c0.hi, c1.lo, c1.hi };

  // buf[j&1][k][tid] element-major: DS writes/reads are bank-conflict free.
  unsigned lbase = gbuf_base + (unsigned)((((j & 1) * 16) * TPB + tid) * 4);
#pragma unroll
  for (int a = 0; a < 4; ++a) {
#pragma unroll
    for (int b = 0; b < 4; ++b) {
      async_gather_b32(lbase + (unsigned)(((a * 4 + b) * TPB) * 4),
                       basec + rofs[a] + cofs[b]);
    }
  }
  return st;
}

__global__ __launch_bounds__(TPB)
void roi_align_fpn_kernel(
    const float* __restrict__ f3, const float* __restrict__ f4,
    const float* __restrict__ f5, const float* __restrict__ f6,
    const float* __restrict__ boxes, const int* __restrict__ img_ids,
    float* __restrict__ out, int C)
{
  __shared__ SampTab rowTab[2 * OUTS];
  __shared__ SampTab colTab[2 * OUTS];
  __shared__ float   gbuf[2 * 16 * TPB];   // 32 KB double buffer

  const int roi = blockIdx.x;
  const int tid = threadIdx.x;

  // ---- per-ROI setup (redundant per thread; cheap) ----
  const float by = boxes[roi * 4 + 0];
  const float bx = boxes[roi * 4 + 1];
  const float bh = boxes[roi * 4 + 2];
  const float bw = boxes[roi * 4 + 3];

  float kf = floorf(4.0f + log2f(sqrtf(bh * bw) * (1.0f / 224.0f)));
  kf = fminf(fmaxf(kf, 3.0f), 6.0f);
  const int   lvl        = (int)kf;
  const float inv_stride = 1.0f / (float)(1 << lvl);

  const float* feat; int H;
  if      (lvl == 3) { feat = f3; H = 128; }
  else if (lvl == 4) { feat = f4; H = 64;  }
  else if (lvl == 5) { feat = f5; H = 32;  }
  else               { feat = f6; H = 16;  }
  const int W  = H;              // all pyramids square
  const int HW = H * W;
  const int img = img_ids[roi];
  const float* base0 = feat + (size_t)img * (size_t)C * (size_t)HW;

  // Reproduce the reference exactly (it feeds tlbr columns into x1,y1,x2,y2):
  // row coordinate (indexes H) runs from l=(bx-bw/2)/s, step max(r-l,1)/7
  // col coordinate (indexes W) runs from t=(by-bh/2)/s, step max(b-t,1)/7
  const float t = (by - bh * 0.5f) * inv_stride;
  const float b = (by + bh * 0.5f) * inv_stride;
  const float l = (bx - bw * 0.5f) * inv_stride;
  const float r = (bx + bw * 0.5f) * inv_stride;
  const float bin_h = fmaxf(r - l, 1.0f) * (1.0f / 7.0f);  // row step
  const float bin_w = fmaxf(b - t, 1.0f) * (1.0f / 7.0f);  // col step

  // ---- separable sample tables: 14 row samples + 14 col samples ----
  if (tid < 4 * OUTS) {
    const bool  isRow = tid < 2 * OUTS;
    const int   s     = isRow ? tid : tid - 2 * OUTS;
    const float off   = (float)(s >> 1) + ((float)(s & 1) + 0.5f) * 0.5f;
    const float coord = (isRow ? l : t) + off * (isRow ? bin_h : bin_w);
    const bool  valid = (coord >= -1.0f) && (coord <= (float)H);  // H==W
    float cc = fminf(fmaxf(coord, 0.0f), (float)(H - 1));
    int   lo = (int)floorf(cc);
    int   hi = min(lo + 1, H - 1);
    float fr = cc - (float)lo;
    SampTab e;
    e.lo = lo; e.hi = hi;
    e.wl = valid ? (1.0f - fr) : 0.0f;
    e.wh = valid ? fr          : 0.0f;
    if (isRow) rowTab[s] = e; else colTab[s] = e;
  }
  __syncthreads();

  const unsigned gbuf_base = (unsigned)(unsigned long long)(void*)&gbuf[0]; // low 32b = LDS offset
  const int nIter = (C * NBINS) / TPB;   // 49 for C=256
  float* outr = out + (size_t)roi * (size_t)C * (size_t)NBINS;

  // ---- software pipeline: async gather batch j+1 while reducing batch j ----
  Stage cur = issue_stage(0, tid, base0, HW, W, rowTab, colTab, gbuf_base);
  for (int j = 0; j < nIter; ++j) {
    Stage nxt = cur;
    if (j + 1 < nIter) {
      wait_ds_0();   // DS reads of buf[(j+1)&1] (from iter j-1) done before refill
      nxt = issue_stage(j + 1, tid, base0, HW, W, rowTab, colTab, gbuf_base);
      wait_async_le16();   // <=16 outstanding => batch j fully landed in LDS
    } else {
      wait_async_0();
    }

    const float* gb = &gbuf[((j & 1) * 16) * TPB + tid];
    float v0  = gb[ 0 * TPB], v1  = gb[ 1 * TPB], v2  = gb[ 2 * TPB], v3  = gb[ 3 * TPB];
    float v4  = gb[ 4 * TPB], v5  = gb[ 5 * TPB], v6  = gb[ 6 * TPB], v7  = gb[ 7 * TPB];
    float v8  = gb[ 8 * TPB], v9  = gb[ 9 * TPB], v10 = gb[10 * TPB], v11 = gb[11 * TPB];
    float v12 = gb[12 * TPB], v13 = gb[13 * TPB], v14 = gb[14 * TPB], v15 = gb[15 * TPB];

    float rs0 = cur.wx0 * v0  + cur.wx1 * v1  + cur.wx2 * v2  + cur.wx3 * v3;
    float rs1 = cur.wx0 * v4  + cur.wx1 * v5  + cur.wx2 * v6  + cur.wx3 * v7;
    float rs2 = cur.wx0 * v8  + cur.wx1 * v9  + cur.wx2 * v10 + cur.wx3 * v11;
    float rs3 = cur.wx0 * v12 + cur.wx1 * v13 + cur.wx2 * v14 + cur.wx3 * v15;
    float acc = cur.wy0 * rs0 + cur.wy1 * rs1 + cur.wy2 * rs2 + cur.wy3 * rs3;

    outr[cur.p] = acc * 0.25f;   // mean over 2x2 subsamples (invalids already 0)
    cur = nxt;
  }
}

extern "C" void kernel_launch(void* const* d_in, const int* in_sizes, int n_in,
                              void* d_out, int out_size, void* d_ws, size_t ws_size,
                              hipStream_t stream) {
  const float* f3    = (const float*)d_in[0];
  const float* f4    = (const float*)d_in[1];
  const float* f5    = (const float*)d_in[2];
  const float* f6    = (const float*)d_in[3];
  const float* boxes = (const float*)d_in[4];
  const int*   ids   = (const int*)d_in[5];
  float*       out   = (float*)d_out;

  const int N = in_sizes[4] / 4;                 // 1024 ROIs
  const int C = out_size / (N * NBINS);          // 256 channels

  roi_align_fpn_kernel<<<dim3(N), dim3(TPB), 0, stream>>>(
      f3, f4, f5, f6, boxes, ids, out, C);
}